// GATLayer_19688130085867
// MI455X (gfx1250) — compile-verified
//
#include <hip/hip_runtime.h>
#include <hip/hip_bf16.h>

#define IN_DIM 256
#define NCOL 128          // HEADS * OUT_DIM
#define HEADS 4
#define OUT_DIM 32
#define ALPHA 0.2f

typedef __attribute__((ext_vector_type(2))) float v2f;
typedef __attribute__((ext_vector_type(8))) float v8f;

// ---------------------------------------------------------------------------
// Kernel 1: ft = feature @ W_fc   via V_WMMA_F32_16X16X4_F32
// A: M x 256 (row major), B: 256 x 128 (row major), C: M x 128
// One 16x16 output tile per wave, K accumulated in steps of 4.
// ---------------------------------------------------------------------------
__global__ void gat_gemm_wmma(const float* __restrict__ A,
                              const float* __restrict__ B,
                              float* __restrict__ C, int M) {
    const int lane = threadIdx.x & 31;
    const int wave = (blockIdx.x * blockDim.x + threadIdx.x) >> 5;
    const int tiles_n = NCOL / 16;            // 8
    const int row_tile = wave / tiles_n;
    const int col_tile = wave % tiles_n;
    const int m0 = row_tile * 16;
    if (m0 >= M) return;                      // wave-uniform: EXEC stays all-1s

    const int n0   = col_tile * 16;
    const int r    = lane & 15;               // row (A) / col (B) within tile
    const int koff = (lane >> 4) * 2;         // lanes 16-31 hold K+2,K+3

    v8f acc = {};
    const float* arow = A + (size_t)(m0 + r) * IN_DIM;
#pragma unroll 4
    for (int k = 0; k < IN_DIM; k += 4) {
        // A frag: {A[m0+r][k+koff], A[m0+r][k+koff+1]}  (8B aligned -> b64 load)
        v2f a = *(const v2f*)(arow + k + koff);
        // B frag: {B[k+koff][n0+r], B[k+koff+1][n0+r]}
        const float* bp = B + (size_t)(k + koff) * NCOL + n0 + r;
        v2f b;
        b.x = bp[0];
        b.y = bp[NCOL];
        acc = __builtin_amdgcn_wmma_f32_16x16x4_f32(
            /*neg_a=*/false, a, /*neg_b=*/false, b,
            /*c_mod=*/(short)0, acc, /*reuse_a=*/false, /*reuse_b=*/false);
    }

    // C/D layout: VGPR i -> row m0 + i + 8*(lane>=16), col n0 + (lane&15)
    const int row_add = (lane >> 4) * 8;
    float* cp = C + (size_t)(m0 + row_add) * NCOL + n0 + (lane & 15);
#pragma unroll
    for (int i = 0; i < 8; ++i) cp[(size_t)i * NCOL] = acc[i];
}

// ---------------------------------------------------------------------------
// Kernel 2: per-node attention scores a1[n,h], a2[n,h]
// ---------------------------------------------------------------------------
__global__ void gat_node_scores(const float* __restrict__ ft,
                                const float* __restrict__ attn_l,
                                const float* __restrict__ attn_r,
                                float* __restrict__ a1,
                                float* __restrict__ a2, int n_nodes) {
    int i = blockIdx.x * blockDim.x + threadIdx.x;   // node*HEADS + h
    if (i >= n_nodes * HEADS) return;
    int h = i & (HEADS - 1);
    const float* f  = ft + (size_t)(i >> 2) * NCOL + h * OUT_DIM;
    const float* al = attn_l + h * OUT_DIM;
    const float* ar = attn_r + h * OUT_DIM;
    float s1 = 0.f, s2 = 0.f;
#pragma unroll
    for (int d = 0; d < OUT_DIM; ++d) {
        float v = f[d];
        s1 += v * al[d];
        s2 += v * ar[d];
    }
    a1[i] = s1;
    a2[i] = s2;
}

// ---------------------------------------------------------------------------
// Kernel 3: init out=0, e_sum=0, e_max=-FLT_MAX (buffers are poisoned 0xAA)
// ---------------------------------------------------------------------------
__global__ void gat_init(float* __restrict__ out, float* __restrict__ e_max,
                         float* __restrict__ e_sum, int n_nodes) {
    int i = blockIdx.x * blockDim.x + threadIdx.x;
    int n_out = n_nodes * NCOL;
    if (i < n_out) out[i] = 0.f;
    if (i < n_nodes * HEADS) {
        e_max[i] = -3.402823466e+38f;
        e_sum[i] = 0.f;
    }
}

// float atomic max via signed/unsigned int monotonicity trick
__device__ __forceinline__ void atomicMaxF32(float* addr, float val) {
    if (val >= 0.f)
        atomicMax((int*)addr, __float_as_int(val));
    else
        atomicMin((unsigned int*)addr, __float_as_uint(val));
}

__device__ __forceinline__ float leaky(float v) {
    return v > 0.f ? v : ALPHA * v;
}

// ---------------------------------------------------------------------------
// Kernel 4: segment max of leaky-relu edge logits into e_max[dst,h]
// One thread per edge; a1/a2 gathered as b128 (all 4 heads at once).
// ---------------------------------------------------------------------------
__global__ void gat_edge_max(const int* __restrict__ src,
                             const int* __restrict__ dst,
                             const float* __restrict__ a1,
                             const float* __restrict__ a2,
                             float* __restrict__ e_max, int n_edges) {
    int e = blockIdx.x * blockDim.x + threadIdx.x;
    if (e >= n_edges) return;
    int s = src[e], d = dst[e];
    float4 A1 = *(const float4*)(a1 + (size_t)s * HEADS);
    float4 A2 = *(const float4*)(a2 + (size_t)d * HEADS);
    float* em = e_max + (size_t)d * HEADS;
    atomicMaxF32(em + 0, leaky(A1.x + A2.x));
    atomicMaxF32(em + 1, leaky(A1.y + A2.y));
    atomicMaxF32(em + 2, leaky(A1.z + A2.z));
    atomicMaxF32(em + 3, leaky(A1.w + A2.w));
}

// ---------------------------------------------------------------------------
// Kernel 5: segment sum of exp(e - max) into e_sum[dst,h]
// One thread per edge; b128 gathers for scores and maxima.
// ---------------------------------------------------------------------------
__global__ void gat_edge_sum(const int* __restrict__ src,
                             const int* __restrict__ dst,
                             const float* __restrict__ a1,
                             const float* __restrict__ a2,
                             const float* __restrict__ e_max,
                             float* __restrict__ e_sum, int n_edges) {
    int e = blockIdx.x * blockDim.x + threadIdx.x;
    if (e >= n_edges) return;
    int s = src[e], d = dst[e];
    float4 A1 = *(const float4*)(a1 + (size_t)s * HEADS);
    float4 A2 = *(const float4*)(a2 + (size_t)d * HEADS);
    float4 EM = *(const float4*)(e_max + (size_t)d * HEADS);
    float* es = e_sum + (size_t)d * HEADS;
    atomicAdd(es + 0, expf(leaky(A1.x + A2.x) - EM.x));
    atomicAdd(es + 1, expf(leaky(A1.y + A2.y) - EM.y));
    atomicAdd(es + 2, expf(leaky(A1.z + A2.z) - EM.z));
    atomicAdd(es + 3, expf(leaky(A1.w + A2.w) - EM.w));
}

// ---------------------------------------------------------------------------
// Kernel 6: out[dst] += ft[src] * alpha
// 32 threads per edge, float4 (b128) per thread; alpha recomputed per thread
// (8 lanes share each score address -> coalesced broadcast loads).
// ---------------------------------------------------------------------------
__global__ void gat_edge_agg(const int* __restrict__ src,
                             const int* __restrict__ dst,
                             const float* __restrict__ ft,
                             const float* __restrict__ a1,
                             const float* __restrict__ a2,
                             const float* __restrict__ e_max,
                             const float* __restrict__ e_sum,
                             float* __restrict__ out, int n_edges) {
    long long t = (long long)blockIdx.x * blockDim.x + threadIdx.x;
    if (t >= (long long)n_edges * 32) return;
    int e = (int)(t >> 5);         // edge
    int q = (int)(t & 31);         // quad index: dims 4q..4q+3
    int h = q >> 3;                // head = (4q)/32
    int s = src[e], d = dst[e];
    int idx = d * HEADS + h;
    float v = leaky(a1[s * HEADS + h] + a2[idx]);
    float alpha = expf(v - e_max[idx]) / e_sum[idx];
    float4 F = *(const float4*)(ft + (size_t)s * NCOL + 4 * q);
    float* op = out + (size_t)d * NCOL + 4 * q;
    atomicAdd(op + 0, F.x * alpha);
    atomicAdd(op + 1, F.y * alpha);
    atomicAdd(op + 2, F.z * alpha);
    atomicAdd(op + 3, F.w * alpha);
}

// ---------------------------------------------------------------------------
extern "C" void kernel_launch(void* const* d_in, const int* in_sizes, int n_in,
                              void* d_out, int out_size, void* d_ws, size_t ws_size,
                              hipStream_t stream) {
    const float* feature = (const float*)d_in[0];
    const int*   src     = (const int*)d_in[1];
    const int*   dst     = (const int*)d_in[2];
    const float* W_fc    = (const float*)d_in[3];
    const float* attn_l  = (const float*)d_in[4];
    const float* attn_r  = (const float*)d_in[5];
    float*       out     = (float*)d_out;

    const int M = in_sizes[0] / IN_DIM;     // 100000 nodes
    const int E = in_sizes[1];              // 1600000 edges

    // workspace layout: ft [M*128] | a1 [M*4] | a2 [M*4] | e_max [M*4] | e_sum [M*4]
    float* ft    = (float*)d_ws;
    float* a1    = ft    + (size_t)M * NCOL;
    float* a2    = a1    + (size_t)M * HEADS;
    float* e_max = a2    + (size_t)M * HEADS;
    float* e_sum = e_max + (size_t)M * HEADS;

    // 1) projection GEMM (WMMA), one 16x16 tile per wave, 8 waves per block
    {
        int n_waves = ((M + 15) / 16) * (NCOL / 16);          // 50000
        int threads = n_waves * 32;
        gat_gemm_wmma<<<(threads + 255) / 256, 256, 0, stream>>>(feature, W_fc, ft, M);
    }
    // 2) per-node scores
    gat_node_scores<<<(M * HEADS + 255) / 256, 256, 0, stream>>>(ft, attn_l, attn_r,
                                                                 a1, a2, M);
    // 3) init accumulators / output
    gat_init<<<(M * NCOL + 255) / 256, 256, 0, stream>>>(out, e_max, e_sum, M);
    // 4) segment max (1 thread / edge)
    gat_edge_max<<<(E + 255) / 256, 256, 0, stream>>>(src, dst, a1, a2, e_max, E);
    // 5) segment exp-sum (1 thread / edge)
    gat_edge_sum<<<(E + 255) / 256, 256, 0, stream>>>(src, dst, a1, a2,
                                                      e_max, e_sum, E);
    // 6) weighted scatter aggregation (32 threads / edge, float4 each)
    {
        long long total = (long long)E * 32;
        int blocks = (int)((total + 255) / 256);
        gat_edge_agg<<<blocks, 256, 0, stream>>>(src, dst, ft, a1, a2,
                                                 e_max, e_sum, out, E);
    }
}